// TransformRPNOutputs_44693429682209
// MI455X (gfx1250) — compile-verified
//
#include <hip/hip_runtime.h>
#include <cstdint>
#include <cstddef>

typedef unsigned int u32;
typedef unsigned long long u64;
typedef float v2f __attribute__((ext_vector_type(2)));
typedef float v8f __attribute__((ext_vector_type(8)));

#define NUM_IMG   4
#define M_CAND    6960
#define COL_WORDS 218      // ceil(6960/32)
#define ROW_TILES 435      // 6960/16
#define SORT_N    8192
#define PRE_NMS   2000
#define MAX_OUT   1000
#define NMS_THR_F 0.7f
#define MAX_SHAPE_F 512.0f
#define MAX_RATIO_F 4.135166556742356f

// per-level constants
__constant__ int   d_W[5]      = {128, 64, 32, 16, 8};
__constant__ int   d_H[5]      = {128, 64, 32, 16, 8};
__constant__ int   d_STRIDE[5] = {4, 8, 16, 32, 64};
__constant__ int   d_BASE[5]   = {0, 2000, 4000, 6000, 6768};
__constant__ int   d_KOFF[3]   = {0, 49152, 61440};
__constant__ float d_WR[3]     = {1.41421356237f, 1.0f, 0.70710678118f};
__constant__ float d_HR[3]     = {0.70710678118f, 1.0f, 1.41421356237f};

#define KEYS_PER_B 64512   // levels 0..2
#define ALL_PER_B  65472   // levels 0..4

// ---------------------------------------------------------------- init
__global__ void init_kernel(u32* prefix, u32* kRem, u32* cntGt, u32* cntEq,
                            u32* hist, int* keepCount,
                            u32* candKey, int* candIdx, int* candLev) {
  int t = blockIdx.x * blockDim.x + threadIdx.x;
  if (t < 12) { prefix[t] = 0; kRem[t] = PRE_NMS; cntGt[t] = 0; cntEq[t] = 0; }
  if (t < 12 * 256) hist[t] = 0;
  if (t < NUM_IMG) keepCount[t] = 0;
  if (t < NUM_IMG * M_CAND) { candKey[t] = 0; candIdx[t] = 0; candLev[t] = 0; }
}

// ---------------------------------------------------------------- scores
__global__ void score_kernel(const float* __restrict__ c0, const float* __restrict__ c1,
                             const float* __restrict__ c2, const float* __restrict__ c3,
                             const float* __restrict__ c4,
                             u32* __restrict__ keys,
                             u32* __restrict__ candKey, int* __restrict__ candIdx,
                             int* __restrict__ candLev) {
  int t = blockIdx.x * blockDim.x + threadIdx.x;
  if (t >= NUM_IMG * ALL_PER_B) return;
  int b = t / ALL_PER_B;
  int r = t - b * ALL_PER_B;
  int l, i;
  if      (r < 49152) { l = 0; i = r; }
  else if (r < 61440) { l = 1; i = r - 49152; }
  else if (r < 64512) { l = 2; i = r - 61440; }
  else if (r < 65280) { l = 3; i = r - 64512; }
  else                { l = 4; i = r - 65280; }
  const float* cls = (l == 0) ? c0 : (l == 1) ? c1 : (l == 2) ? c2 : (l == 3) ? c3 : c4;
  int W = d_W[l], H = d_H[l];
  int a = i % 3, cell = i / 3;
  int x = cell % W, y = cell / W;
  float v = cls[((b * 3 + a) * H + y) * W + x];
  float sc = 1.0f / (1.0f + __expf(-v));     // sigmoid, in (0,1): bits are order-preserving
  u32 key = __float_as_uint(sc);
  if (l < 3) {
    keys[b * KEYS_PER_B + d_KOFF[l] + i] = key;
  } else {
    int slot = b * M_CAND + d_BASE[l] + i;   // levels 3/4: take everything
    candKey[slot] = key; candIdx[slot] = i; candLev[slot] = l;
  }
}

// ---------------------------------------------------------------- radix select (top-2000 exact key threshold)
__global__ void hist_kernel(const u32* __restrict__ keys, const u32* __restrict__ prefix,
                            u32* __restrict__ hist, int pass) {
  int t = blockIdx.x * blockDim.x + threadIdx.x;
  if (t >= NUM_IMG * KEYS_PER_B) return;
  int b = t / KEYS_PER_B;
  int r = t - b * KEYS_PER_B;
  int l = (r < 49152) ? 0 : (r < 61440) ? 1 : 2;
  u32 key = keys[t];
  int shift = 24 - 8 * pass;
  bool match = (pass == 0) || ((key >> (shift + 8)) == prefix[b * 3 + l]);
  if (match) atomicAdd(&hist[(b * 3 + l) * 256 + ((key >> shift) & 255u)], 1u);
}

__global__ void scan_kernel(u32* prefix, u32* kRem, u32* hist) {
  int t = threadIdx.x;
  if (t >= 12) return;
  u32 cum = 0, kr = kRem[t], pf = prefix[t];
  int chosen = 0;
  for (int d = 255; d >= 0; --d) {
    u32 c = hist[t * 256 + d];
    if (cum + c >= kr) { chosen = d; break; }
    cum += c;
  }
  prefix[t] = (pf << 8) | (u32)chosen;
  kRem[t] = kr - cum;
  for (int d = 0; d < 256; ++d) hist[t * 256 + d] = 0;
}

__global__ void select_kernel(const u32* __restrict__ keys, const u32* __restrict__ prefix,
                              const u32* __restrict__ kRem, u32* cntGt, u32* cntEq,
                              u32* __restrict__ candKey, int* __restrict__ candIdx,
                              int* __restrict__ candLev) {
  int t = blockIdx.x * blockDim.x + threadIdx.x;
  if (t >= NUM_IMG * KEYS_PER_B) return;
  int b = t / KEYS_PER_B;
  int r = t - b * KEYS_PER_B;
  int l = (r < 49152) ? 0 : (r < 61440) ? 1 : 2;
  int i = r - d_KOFF[l];
  u32 key = keys[t];
  int pl = b * 3 + l;
  u32 T = prefix[pl], kr = kRem[pl];
  int slot = -1;
  if (key > T) {
    u32 p = atomicAdd(&cntGt[pl], 1u);
    slot = d_BASE[l] + (int)p;
  } else if (key == T) {
    u32 p = atomicAdd(&cntEq[pl], 1u);
    if (p < kr) slot = d_BASE[l] + (PRE_NMS - (int)kr) + (int)p;
  }
  if (slot >= 0) {
    int g = b * M_CAND + slot;
    candKey[g] = key; candIdx[g] = i; candLev[g] = l;
  }
}

// ---------------------------------------------------------------- per-image sort, fully in LDS (64KB of 320KB/WGP)
__global__ __launch_bounds__(1024) void sort_kernel(const u32* __restrict__ candKey,
                                                    const int* __restrict__ candIdx,
                                                    const int* __restrict__ candLev,
                                                    u32* __restrict__ sKey,
                                                    int* __restrict__ sLev,
                                                    int* __restrict__ sIdx) {
  __shared__ u64 items[SORT_N];
  int b = blockIdx.x, tid = threadIdx.x;
  for (int i = tid; i < SORT_N; i += 1024) {
    u64 it = 0;
    if (i < M_CAND) {
      u32 key = candKey[b * M_CAND + i];
      u32 sec = ((u32)candLev[b * M_CAND + i] << 17) | (u32)candIdx[b * M_CAND + i];
      it = ((u64)key << 32) | (u64)(0xFFFFFFFFu - sec);  // tie-break: level asc, idx asc
    }
    items[i] = it;
  }
  __syncthreads();
  for (int k = 2; k <= SORT_N; k <<= 1) {
    for (int j = k >> 1; j > 0; j >>= 1) {
      for (int i = tid; i < SORT_N; i += 1024) {
        int ixj = i ^ j;
        if (ixj > i) {
          u64 a = items[i], c = items[ixj];
          bool up = ((i & k) == 0);               // descending overall
          if ((a < c) == up) { items[i] = c; items[ixj] = a; }
        }
      }
      __syncthreads();
    }
  }
  for (int r = tid; r < M_CAND; r += 1024) {
    u64 it = items[r];
    u32 sec = 0xFFFFFFFFu - (u32)(it & 0xFFFFFFFFu);
    sKey[b * M_CAND + r] = (u32)(it >> 32);
    sLev[b * M_CAND + r] = (int)(sec >> 17);
    sIdx[b * M_CAND + r] = (int)(sec & 0x1FFFFu);
  }
}

// ---------------------------------------------------------------- decode (anchors computed analytically)
__global__ void decode_kernel(const float* __restrict__ r0, const float* __restrict__ r1,
                              const float* __restrict__ r2, const float* __restrict__ r3,
                              const float* __restrict__ r4,
                              const u32* __restrict__ sKey, const int* __restrict__ sLev,
                              const int* __restrict__ sIdx,
                              float4* __restrict__ sBox, float* __restrict__ sScore,
                              u32* __restrict__ sValid) {
  int t = blockIdx.x * blockDim.x + threadIdx.x;
  if (t >= NUM_IMG * M_CAND) return;
  int b = t / M_CAND;
  int lev = sLev[t], idx = sIdx[t];
  u32 key = sKey[t];
  const float* reg = (lev == 0) ? r0 : (lev == 1) ? r1 : (lev == 2) ? r2 : (lev == 3) ? r3 : r4;
  int W = d_W[lev], H = d_H[lev];
  float stride = (float)d_STRIDE[lev];
  int a = idx % 3, cell = idx / 3;
  int x = cell % W, y = cell / W;
  int hw = H * W;
  int base = ((b * 12 + a * 4) * H + y) * W + x;
  float dx = reg[base], dy = reg[base + hw];
  float dw = reg[base + 2 * hw], dh = reg[base + 3 * hw];
  float pw = stride * 8.0f * d_WR[a], ph = stride * 8.0f * d_HR[a];
  float px = (float)x * stride, py = (float)y * stride;
  dw = fminf(fmaxf(dw, -MAX_RATIO_F), MAX_RATIO_F);
  dh = fminf(fmaxf(dh, -MAX_RATIO_F), MAX_RATIO_F);
  float gw = pw * __expf(dw), gh = ph * __expf(dh);
  float gx = px + pw * dx, gy = py + ph * dy;
  float x1 = fminf(fmaxf(gx - gw * 0.5f, 0.f), MAX_SHAPE_F);
  float y1 = fminf(fmaxf(gy - gh * 0.5f, 0.f), MAX_SHAPE_F);
  float x2 = fminf(fmaxf(gx + gw * 0.5f, 0.f), MAX_SHAPE_F);
  float y2 = fminf(fmaxf(gy + gh * 0.5f, 0.f), MAX_SHAPE_F);
  sBox[t] = make_float4(x1, y1, x2, y2);
  sScore[t] = __uint_as_float(key);
  sValid[t] = (((x2 - x1) > 0.f) && ((y2 - y1) > 0.f)) ? 1u : 0u;
}

// ---------------------------------------------------------------- NMS suppression-mask: wave32 tile of 16 rows x 32 cols.
// V_WMMA_F32_16X16X4_F32 computes area_i + area_j for the whole 16x16 tile
// (rank-2 outer product: A=[area_r, 1], B=[1, area_c]) on the matrix pipe,
// co-executing with the branchless VALU overlap math. Box tiles staged via
// async global->LDS DMA; per-lane row boxes hoisted to registers so the
// inner loop is pure VALU + ballot (no EXEC branches, no redundant ds_loads).
__global__ __launch_bounds__(32) void mask_kernel(const float4* __restrict__ sBox,
                                                  const int* __restrict__ sLev,
                                                  u32* __restrict__ mask) {
  const int b = blockIdx.z;
  const int rowBase = blockIdx.y * 16;
  const int colBase = blockIdx.x * 32;
  if (colBase + 31 <= rowBase) return;   // uniform: strictly-lower-triangle block
  const int lane = threadIdx.x;

  __shared__ float4 colBox[32];
  __shared__ float4 rowBox[16];
  __shared__ int    colLv[32];
  __shared__ int    rowLv[16];

  {
    int c = colBase + lane; if (c >= M_CAND) c = M_CAND - 1;
    u64 gb = (u64)(uintptr_t)(sBox + (size_t)b * M_CAND + c);
    u32 lb = (u32)(uintptr_t)(&colBox[lane]);
    asm volatile("global_load_async_to_lds_b128 %0, %1, off" :: "v"(lb), "v"(gb) : "memory");
    u64 gl = (u64)(uintptr_t)(sLev + (size_t)b * M_CAND + c);
    u32 ll = (u32)(uintptr_t)(&colLv[lane]);
    asm volatile("global_load_async_to_lds_b32 %0, %1, off" :: "v"(ll), "v"(gl) : "memory");
    if (lane < 16) {
      int rr = rowBase + lane; if (rr >= M_CAND) rr = M_CAND - 1;
      u64 gr = (u64)(uintptr_t)(sBox + (size_t)b * M_CAND + rr);
      u32 lr = (u32)(uintptr_t)(&rowBox[lane]);
      asm volatile("global_load_async_to_lds_b128 %0, %1, off" :: "v"(lr), "v"(gr) : "memory");
      u64 grl = (u64)(uintptr_t)(sLev + (size_t)b * M_CAND + rr);
      u32 lrl = (u32)(uintptr_t)(&rowLv[lane]);
      asm volatile("global_load_async_to_lds_b32 %0, %1, off" :: "v"(lrl), "v"(grl) : "memory");
    }
    asm volatile("s_wait_asynccnt 0" ::: "memory");
  }
  __syncthreads();

  // A-matrix (16x4): lane<16 holds M=lane, K0=area_row, K1=1; lanes 16-31 (K2,K3) zero.
  v2f A; A.x = 0.f; A.y = 0.f;
  if (lane < 16) {
    float4 rbl = rowBox[lane];
    A.x = (rbl.z - rbl.x) * (rbl.w - rbl.y);
    A.y = 1.0f;
  }

  // Hoist this lane's 8 row boxes + levels into registers (one ds burst, no
  // LDS traffic inside the pair loop).
  const bool hiHalf = (lane >= 16);
  const int roff = hiHalf ? 8 : 0;
  float4 rb[8];
  int    rlv[8];
  #pragma unroll
  for (int v = 0; v < 8; ++v) {
    rb[v]  = rowBox[v + roff];
    rlv[v] = rowLv[v + roff];
  }

  u32 wlo[8], whi[8];
  const int cloc0 = lane & 15;

  #pragma unroll
  for (int tile = 0; tile < 2; ++tile) {
    // B-matrix (4x16): lane<16 holds N=lane, K0=1, K1=area_col; lanes 16-31 zero.
    v2f B; B.x = 0.f; B.y = 0.f;
    if (lane < 16) {
      float4 cb2 = colBox[tile * 16 + lane];
      B.x = 1.0f;
      B.y = (cb2.z - cb2.x) * (cb2.w - cb2.y);
    }
    v8f acc = {0.f, 0.f, 0.f, 0.f, 0.f, 0.f, 0.f, 0.f};
    acc = __builtin_amdgcn_wmma_f32_16x16x4_f32(false, A, false, B, (short)0, acc,
                                                false, false);
    // D layout: lane holds (m = v + 8*(lane>=16), n = lane&15), acc[v] = area_m + area_n
    const int cloc = tile * 16 + cloc0;
    const int col = colBase + cloc;
    const float4 cb = colBox[cloc];
    const int clv = colLv[cloc];
    const int colOk = (int)(col < M_CAND);
    #pragma unroll
    for (int v = 0; v < 8; ++v) {
      int row = rowBase + v + roff;
      float iw = fmaxf(fminf(rb[v].z, cb.z) - fmaxf(rb[v].x, cb.x), 0.f);
      float ih = fmaxf(fminf(rb[v].w, cb.w) - fmaxf(rb[v].y, cb.y), 0.f);
      float inter = iw * ih;
      float uni = acc[v] - inter;            // area_r + area_c - inter (from WMMA)
      // branchless: bitwise-& of comparison results, no short-circuit branches
      int p = (int)(rlv[v] == clv) & (int)(col > row) & colOk &
              (int)(uni > 0.f) & (int)(inter > NMS_THR_F * uni);
      u32 bal = (u32)__ballot(p);
      if (tile == 0) { wlo[v] = bal & 0xFFFFu;           whi[v] = bal >> 16; }
      else           { wlo[v] |= (bal & 0xFFFFu) << 16;  whi[v] |= (bal & 0xFFFF0000u); }
    }
  }
  if (lane == 0) {
    int cw = colBase >> 5;
    #pragma unroll
    for (int v = 0; v < 8; ++v) {
      mask[((size_t)b * M_CAND + rowBase + v) * COL_WORDS + cw] = wlo[v];
      mask[((size_t)b * M_CAND + rowBase + v + 8) * COL_WORDS + cw] = whi[v];
    }
  }
}

// ---------------------------------------------------------------- greedy reduce (1 block / image)
__global__ __launch_bounds__(256) void reduce_kernel(const u32* __restrict__ mask,
                                                     const u32* __restrict__ sValid,
                                                     int* __restrict__ keepList,
                                                     int* __restrict__ keepCount) {
  int b = blockIdx.x, tid = threadIdx.x;
  __shared__ u32 removed[COL_WORDS];
  __shared__ int sFlag;
  __shared__ int sNum;
  for (int w = tid; w < COL_WORDS; w += 256) removed[w] = 0;
  if (tid == 0) sNum = 0;
  __syncthreads();
  for (int i = 0; i < M_CAND; ++i) {
    if (tid == 0) {
      int alive = (sValid[b * M_CAND + i] != 0u) &&
                  (((removed[i >> 5] >> (i & 31)) & 1u) == 0u);
      sFlag = alive;
      if (alive) keepList[b * M_CAND + sNum++] = i;
    }
    __syncthreads();
    if (sFlag) {
      const u32* mrow = mask + ((size_t)b * M_CAND + i) * COL_WORDS;
      for (int w = (i >> 5) + tid; w < COL_WORDS; w += 256) removed[w] |= mrow[w];
    }
    __syncthreads();
  }
  if (tid == 0) keepCount[b] = sNum;
}

// ---------------------------------------------------------------- output
__global__ void output_kernel(const float4* __restrict__ sBox, const float* __restrict__ sScore,
                              const int* __restrict__ keepList, const int* __restrict__ keepCount,
                              float* __restrict__ out) {
  int t = blockIdx.x * blockDim.x + threadIdx.x;
  if (t >= NUM_IMG * MAX_OUT) return;
  int b = t / MAX_OUT, r = t % MAX_OUT;
  float* o = out + (size_t)t * 5;
  if (r < keepCount[b]) {
    int i = keepList[b * M_CAND + r];
    float4 bx = sBox[b * M_CAND + i];
    o[0] = bx.x; o[1] = bx.y; o[2] = bx.z; o[3] = bx.w;
    o[4] = sScore[b * M_CAND + i];
  } else {
    o[0] = 0.f; o[1] = 0.f; o[2] = 0.f; o[3] = 0.f; o[4] = 0.f;
  }
}

// ---------------------------------------------------------------- host
extern "C" void kernel_launch(void* const* d_in, const int* in_sizes, int n_in,
                              void* d_out, int out_size, void* d_ws, size_t ws_size,
                              hipStream_t stream) {
  const float* cls[5];
  const float* reg[5];
  bool interleaved = (n_in >= 2) && (in_sizes[1] > in_sizes[0]);  // reg0 is 4x cls0
  for (int l = 0; l < 5; ++l) {
    if (interleaved) { cls[l] = (const float*)d_in[2 * l]; reg[l] = (const float*)d_in[2 * l + 1]; }
    else             { cls[l] = (const float*)d_in[l];     reg[l] = (const float*)d_in[5 + l]; }
  }

  char* p = (char*)d_ws;
  auto take = [&](size_t bytes) -> void* {
    void* r = (void*)p;
    p += (bytes + 255) & ~(size_t)255;
    return r;
  };
  u32*  keys      = (u32*)take((size_t)NUM_IMG * KEYS_PER_B * 4);
  u32*  candKey   = (u32*)take((size_t)NUM_IMG * M_CAND * 4);
  int*  candIdx   = (int*)take((size_t)NUM_IMG * M_CAND * 4);
  int*  candLev   = (int*)take((size_t)NUM_IMG * M_CAND * 4);
  u32*  prefix    = (u32*)take(12 * 4);
  u32*  kRem      = (u32*)take(12 * 4);
  u32*  cntGt     = (u32*)take(12 * 4);
  u32*  cntEq     = (u32*)take(12 * 4);
  u32*  hist      = (u32*)take(12 * 256 * 4);
  u32*  sKey      = (u32*)take((size_t)NUM_IMG * M_CAND * 4);
  int*  sLev      = (int*)take((size_t)NUM_IMG * M_CAND * 4);
  int*  sIdx      = (int*)take((size_t)NUM_IMG * M_CAND * 4);
  float4* sBox    = (float4*)take((size_t)NUM_IMG * M_CAND * 16);
  float* sScore   = (float*)take((size_t)NUM_IMG * M_CAND * 4);
  u32*  sValid    = (u32*)take((size_t)NUM_IMG * M_CAND * 4);
  u32*  maskBuf   = (u32*)take((size_t)NUM_IMG * M_CAND * COL_WORDS * 4);
  int*  keepList  = (int*)take((size_t)NUM_IMG * M_CAND * 4);
  int*  keepCount = (int*)take(NUM_IMG * 4);
  (void)ws_size; (void)out_size;

  const int initN = NUM_IMG * M_CAND;
  init_kernel<<<(initN + 255) / 256, 256, 0, stream>>>(prefix, kRem, cntGt, cntEq, hist,
                                                       keepCount, candKey, candIdx, candLev);

  const int scoreN = NUM_IMG * ALL_PER_B;
  score_kernel<<<(scoreN + 255) / 256, 256, 0, stream>>>(cls[0], cls[1], cls[2], cls[3], cls[4],
                                                         keys, candKey, candIdx, candLev);

  const int keyN = NUM_IMG * KEYS_PER_B;
  for (int pass = 0; pass < 4; ++pass) {
    hist_kernel<<<(keyN + 255) / 256, 256, 0, stream>>>(keys, prefix, hist, pass);
    scan_kernel<<<1, 32, 0, stream>>>(prefix, kRem, hist);
  }
  select_kernel<<<(keyN + 255) / 256, 256, 0, stream>>>(keys, prefix, kRem, cntGt, cntEq,
                                                        candKey, candIdx, candLev);

  sort_kernel<<<NUM_IMG, 1024, 0, stream>>>(candKey, candIdx, candLev, sKey, sLev, sIdx);

  decode_kernel<<<(initN + 255) / 256, 256, 0, stream>>>(reg[0], reg[1], reg[2], reg[3], reg[4],
                                                         sKey, sLev, sIdx, sBox, sScore, sValid);

  dim3 mgrid(COL_WORDS, ROW_TILES, NUM_IMG);
  mask_kernel<<<mgrid, 32, 0, stream>>>(sBox, sLev, maskBuf);

  reduce_kernel<<<NUM_IMG, 256, 0, stream>>>(maskBuf, sValid, keepList, keepCount);

  const int outN = NUM_IMG * MAX_OUT;
  output_kernel<<<(outN + 255) / 256, 256, 0, stream>>>(sBox, sScore, keepList, keepCount,
                                                        (float*)d_out);
}